// Atoms_18957985644849
// MI455X (gfx1250) — compile-verified
//
#include <hip/hip_runtime.h>
#include <math.h>

#define NS 32768
#define HALF_NS 16384
#define NF 128
#define NH 8
#define NB 16
#define DIMX 428
#define PI_F 3.14159265358979f
#define TWO_PI_F 6.283185307179586f
#define NYQ (22050.0f*0.5f)
#define MAXF (3000.0f/NYQ)
#define MINF (20.0f/NYQ)
#define FSPAN (MAXF-MINF)

typedef float v2f __attribute__((ext_vector_type(2)));
typedef float v8f __attribute__((ext_vector_type(8)));
typedef unsigned int u32x4 __attribute__((ext_vector_type(4)));
typedef int i32x4 __attribute__((ext_vector_type(4)));
typedef int i32x8 __attribute__((ext_vector_type(8)));

#if defined(__has_builtin)
# if __has_builtin(__builtin_amdgcn_tensor_load_to_lds) && __has_builtin(__builtin_amdgcn_s_wait_tensorcnt)
#  define HAVE_TDM 1
# endif
#endif

__device__ __forceinline__ float sigm(float v) { return 1.0f/(1.0f+expf(-v)); }

// lin_interp semantics identical to the reference
__device__ __forceinline__ float lin_interp_arr(const float* a, int n, int t, int outN) {
    float scale = (float)n/(float)outN;
    float coord = ((float)t + 0.5f)*scale - 0.5f;
    coord = fminf(fmaxf(coord, 0.0f), (float)(n-1));
    int lo = (int)floorf(coord);
    int hi = min(lo+1, n-1);
    float w = coord - (float)lo;
    return a[lo]*(1.0f-w) + a[hi]*w;
}

#ifdef HAVE_TDM
// ---------------------------------------------------------------------------
// Tensor Data Mover: 1-D global->LDS DMA. D# per CDNA5 ISA ch.8:
// group0: count=1, lds_addr, global_addr[56:0], type=2 ("image").
// group1: data_size=4B (code 2), tensor_dim0=n, tensor_dim1=1,
//         tile_dim0=n, tile_dim1=1, dim0_stride=n. 2-group form (<=2D).
// Issue from one wave only (TDM ignores EXEC); wait with s_wait_tensorcnt.
// ---------------------------------------------------------------------------
__device__ __forceinline__ void tdm_load_1d(float* lds_dst, const float* gsrc, int n) {
    unsigned long long ga = (unsigned long long)(uintptr_t)gsrc;
    unsigned int lo = (unsigned int)(uintptr_t)lds_dst;   // LDS byte offset
    u32x4 g0;
    g0[0] = 1u;                                           // count=1, user mode
    g0[1] = lo;                                           // lds_addr (bytes)
    g0[2] = (unsigned int)ga;                             // global_addr[31:0]
    g0[3] = (unsigned int)(ga >> 32) | (2u << 30);        // addr[56:32] | type=2
    i32x8 g1;
    g1[0] = 0x20000;                                      // data_size=2 -> 4 bytes
    g1[1] = (n & 0xffff) << 16;                           // tensor_dim0[15:0]
    g1[2] = ((n >> 16) & 0xffff) | (1 << 16);             // tensor_dim0[31:16] | tensor_dim1=1
    g1[3] = (n & 0xffff) << 16;                           // tile_dim0
    g1[4] = 1;                                            // tile_dim1=1, tile_dim2=0
    g1[5] = n;                                            // tensor_dim0_stride[31:0]
    g1[6] = 0;
    g1[7] = 0;
    i32x4 z4 = {0,0,0,0};
#if __clang_major__ >= 23
    i32x8 z8 = {0,0,0,0,0,0,0,0};
    __builtin_amdgcn_tensor_load_to_lds(g0, g1, z4, z4, z8, 0);
#else
    __builtin_amdgcn_tensor_load_to_lds(g0, g1, z4, z4, 0);
#endif
}
#endif

#define WMMA4(A,B,C) __builtin_amdgcn_wmma_f32_16x16x4_f32(false,(A),false,(B),(short)0,(C),false,false)

// ---------------------------------------------------------------------------
// In-place radix-16 stage (DIF forward / DIT inverse) via V_WMMA_F32_16X16X4_F32
// Butterfly i: j=i%m, blk=i/m, base=blk*n+j, elements at base + r*m.
// Forward: out[u] = W_n^{-ju} * sum_r W16^{-ur} x[r]   (scrambled output)
// Inverse: out[r] = sum_u W16^{+ur} (W_n^{+ju} x[u])   (scrambled input)
// Twiddles are branchless: j==0 gives angle 0 -> identity rotation.
// ---------------------------------------------------------------------------
__device__ void fft_r16_stage(float* re, float* im, int n, bool inv) {
    const int m   = n >> 4;
    const int lgm = 31 - __clz(m);
    const int lane = threadIdx.x & 31;
    const int wave = threadIdx.x >> 5;
    const int nWaves = blockDim.x >> 5;
    const int col = lane & 15;
    const int hi  = lane >> 4;
    const float sgn = inv ? 1.0f : -1.0f;

    // DFT16 matrix chunks (constant across tiles)
    v2f aRe[4], aIm[4], aImN[4];
    #pragma unroll
    for (int kk = 0; kk < 4; ++kk) {
        int k0 = 4*kk + 2*hi;
        float s0,c0,s1,c1;
        __sincosf((float)((col*k0) & 15) * (TWO_PI_F/16.0f), &s0, &c0);
        __sincosf((float)((col*(k0+1)) & 15) * (TWO_PI_F/16.0f), &s1, &c1);
        v2f ar; ar.x = c0;       ar.y = c1;       aRe[kk]  = ar;
        v2f ai; ai.x = sgn*s0;   ai.y = sgn*s1;   aIm[kk]  = ai;
        v2f an; an.x = -sgn*s0;  an.y = -sgn*s1;  aImN[kk] = an;
    }

    const int nTiles = (NS >> 4) >> 4;           // 128
    const float wscale = TWO_PI_F / (float)n;
    for (int t = wave; t < nTiles; t += nWaves) {
        int i    = 16*t + col;                   // butterfly index of this column
        int j    = i & (m-1);
        int blk  = i >> lgm;
        int base = blk*n + j;

        v2f bRe[4], bIm[4];
        #pragma unroll
        for (int kk = 0; kk < 4; ++kk) {
            int r0 = 4*kk + 2*hi;
            int p0 = base + r0*m;
            int p1 = p0 + m;
            float xr0 = re[p0], xi0 = im[p0];
            float xr1 = re[p1], xi1 = im[p1];
            if (inv) {                            // pre-twiddle W_n^{+j*r} (branchless)
                float s,c,tr;
                __sincosf(wscale*(float)(j*r0), &s, &c);
                tr = xr0*c - xi0*s; xi0 = xr0*s + xi0*c; xr0 = tr;
                __sincosf(wscale*(float)(j*(r0+1)), &s, &c);
                tr = xr1*c - xi1*s; xi1 = xr1*s + xi1*c; xr1 = tr;
            }
            v2f br; br.x = xr0; br.y = xr1; bRe[kk] = br;
            v2f bi; bi.x = xi0; bi.y = xi1; bIm[kk] = bi;
        }

        v8f dre = {0.f,0.f,0.f,0.f,0.f,0.f,0.f,0.f};
        v8f dim = {0.f,0.f,0.f,0.f,0.f,0.f,0.f,0.f};
        #pragma unroll
        for (int kk = 0; kk < 4; ++kk) {
            dre = WMMA4(aRe[kk],  bRe[kk], dre);   // Re += Are*Bre
            dre = WMMA4(aImN[kk], bIm[kk], dre);   // Re -= Aim*Bim
            dim = WMMA4(aRe[kk],  bIm[kk], dim);   // Im += Are*Bim
            dim = WMMA4(aIm[kk],  bRe[kk], dim);   // Im += Aim*Bre
        }

        #pragma unroll
        for (int v = 0; v < 8; ++v) {
            int u   = v + 8*hi;
            int pos = base + u*m;
            float orr = dre[v], oii = dim[v];
            if (!inv) {                           // post-twiddle W_n^{-j*u} (branchless)
                float s,c;
                __sincosf(-wscale*(float)(j*u), &s, &c);
                float tr = orr*c - oii*s; oii = orr*s + oii*c; orr = tr;
            }
            re[pos] = orr; im[pos] = oii;
        }
    }
}

// Radix-8 stage (m==1, contiguous, no twiddles). VALU DFT-8 per thread.
__device__ void fft_r8_stage(float* re, float* im, bool inv) {
    const float C8[8] = {1.f, 0.70710678f, 0.f, -0.70710678f, -1.f, -0.70710678f, 0.f, 0.70710678f};
    const float S8[8] = {0.f, 0.70710678f, 1.f, 0.70710678f, 0.f, -0.70710678f, -1.f, -0.70710678f};
    const float sgn = inv ? 1.0f : -1.0f;
    for (int bf = threadIdx.x; bf < NS/8; bf += blockDim.x) {
        int base = bf*8;
        float xr[8], xi[8], yr[8], yi[8];
        #pragma unroll
        for (int r = 0; r < 8; ++r) { xr[r]=re[base+r]; xi[r]=im[base+r]; }
        #pragma unroll
        for (int u = 0; u < 8; ++u) {
            float sr = 0.f, si = 0.f;
            #pragma unroll
            for (int r = 0; r < 8; ++r) {
                int tt = (u*r) & 7;
                float wr = C8[tt], wi = sgn*S8[tt];
                sr += wr*xr[r] - wi*xi[r];
                si += wr*xi[r] + wi*xr[r];
            }
            yr[u] = sr; yi[u] = si;
        }
        #pragma unroll
        for (int u = 0; u < 8; ++u) { re[base+u]=yr[u]; im[base+u]=yi[u]; }
    }
}

// 32768-pt complex FFT, in place. Forward: natural -> digit-scrambled.
// Inverse: scrambled -> natural (unnormalized; caller scales by 1/NS).
__device__ void fft32k(float* re, float* im, bool inv) {
    if (!inv) {
        fft_r16_stage(re, im, 32768, false); __syncthreads();
        fft_r16_stage(re, im,  2048, false); __syncthreads();
        fft_r16_stage(re, im,   128, false); __syncthreads();
        fft_r8_stage (re, im, false);        __syncthreads();
    } else {
        fft_r8_stage (re, im, true);         __syncthreads();
        fft_r16_stage(re, im,   128, true);  __syncthreads();
        fft_r16_stage(re, im,  2048, true);  __syncthreads();
        fft_r16_stage(re, im, 32768, true);  __syncthreads();
    }
}

// True frequency index of scrambled position p (radices 16,16,16,8)
__device__ __forceinline__ int scrambled_to_freq(int p) {
    return (p >> 11) + (((p >> 7) & 15) << 4) + (((p >> 3) & 15) << 8) + ((p & 7) << 12);
}

// ---------------------------------------------------------------------------
// Kernel 1: per-(b,e) -> filtered noise (FFT->shape->IFFT) * gaussian * amps * fe
// ---------------------------------------------------------------------------
extern "C" __global__ void __launch_bounds__(512)
k_atoms(const float* __restrict__ x, const float* __restrict__ noise,
        float* __restrict__ atoms)
{
    extern __shared__ float smem[];
    float* re  = smem;
    float* im  = smem + NS;
    float* aux = smem + 2*NS;     // [0..15]=noise_coeff  [16..143]=fe frames
    const int se = blockIdx.x;
    const float* xp = x + se*DIMX;
    const float* np = noise + se*NS;
    const int tid = threadIdx.x;

#ifdef HAVE_TDM
    if (tid < 32) {                                   // wave 0 issues the DMA
        tdm_load_1d(re, np, NS);
        __builtin_amdgcn_s_wait_tensorcnt(0);
    }
#endif
    if (tid < 16) aux[tid] = sigm(xp[20+tid]);
    if (tid == 0) {
        float run = 0.f;
        for (int i = 0; i < NF; ++i) {
            run += sigm(xp[36+i])*2.0f - 1.0f;
            aux[16+i] = fminf(fmaxf(run, 0.f), 1.f);
        }
    }
#ifdef HAVE_TDM
    __syncthreads();
    for (int t = tid; t < NS; t += blockDim.x) { re[t] = re[t]*2.0f - 1.0f; im[t] = 0.0f; }
#else
    for (int t = tid; t < NS; t += blockDim.x) { re[t] = np[t]*2.0f - 1.0f; im[t] = 0.0f; }
#endif
    __syncthreads();

    fft32k(re, im, false);
    for (int p = tid; p < NS; p += blockDim.x) {
        int k = scrambled_to_freq(p);
        int f = (k <= HALF_NS) ? k : (NS - k);            // symmetric spectrum
        float S = lin_interp_arr(aux, 16, f, HALF_NS + 1);
        re[p] *= S; im[p] *= S;
    }
    __syncthreads();
    fft32k(re, im, true);

    float means = sigm(xp[0]);
    float stds  = sigm(xp[1])*0.1f;
    float amps  = sigm(xp[2]); amps *= amps;
    float mu    = fminf(fmaxf(means*(float)NS, -(float)(NS/2)), (float)NS*1.5f);
    float sigma = fminf(fmaxf((1e-8f + stds)*(float)NS, 0.0f), (float)(NS-1));
    float inv_s = 1.0f/sigma;
    float norm  = inv_s * 0.3989422804014327f;            // 1/(sigma*sqrt(2pi))
    float c1 = fminf(fmaxf(floorf(mu), 0.f), (float)(NS-1));
    float c2 = fminf(c1 + 1.f, (float)(NS-1));
    float z1 = (c1-mu)*inv_s, z2 = (c2-mu)*inv_s;
    float pmax = fmaxf(expf(-0.5f*z1*z1), expf(-0.5f*z2*z2))*norm + 1e-12f;
    float inv_pmax = 1.0f/pmax;
    const float inv_n = 1.0f/(float)NS;

    float* ap = atoms + se*NS;
    for (int t = tid; t < NS; t += blockDim.x) {
        float z = ((float)t - mu)*inv_s;
        float prob = expf(-0.5f*z*z)*norm*inv_pmax;
        float fe = lin_interp_arr(aux+16, NF, t, NS);
        ap[t] = prob * (re[t]*inv_n) * amps * fe;
    }
}

// ---------------------------------------------------------------------------
// Kernel 2: harmonic oscillator bank (phase prefix sum + sin + mag envelope)
// Two-level wave32 shuffle scan replaces the serial scan.
// ---------------------------------------------------------------------------
__device__ __forceinline__ float block_excl_scan_1024(float v, float* wsum) {
    const int lane = threadIdx.x & 31;
    const int wave = threadIdx.x >> 5;
    float x = v;
    #pragma unroll
    for (int d = 1; d < 32; d <<= 1) {
        float y = __shfl_up(x, d, 32);
        if (lane >= d) x += y;
    }                                   // x = inclusive within wave
    if (lane == 31) wsum[wave] = x;
    __syncthreads();
    if (wave == 0) {
        float w = wsum[lane];
        #pragma unroll
        for (int d = 1; d < 32; d <<= 1) {
            float y = __shfl_up(w, d, 32);
            if (lane >= d) w += y;
        }
        wsum[lane] = w;                 // inclusive scan of wave totals
    }
    __syncthreads();
    float waveOff = (wave == 0) ? 0.f : wsum[wave-1];
    return waveOff + (x - v);           // exclusive prefix for this thread
}

extern "C" __global__ void __launch_bounds__(1024)
k_res(const float* __restrict__ x, float* __restrict__ resbuf)
{
    __shared__ float f0fr[NF];
    __shared__ float mgf[NF];
    __shared__ float wsum[32];
    const int se = blockIdx.x;
    const float* xp = x + se*DIMX;
    const int tid = threadIdx.x;

    float f0 = sigm(xp[3]); f0 *= f0;
    if (tid < NF) f0fr[tid] = f0 + sigm(xp[172+tid])*(f0*0.01f);
    __syncthreads();

    const int base = tid*32;
    float acc[32];
    #pragma unroll
    for (int s = 0; s < 32; ++s) acc[s] = 0.f;

    for (int h = 0; h < NH; ++h) {
        float fac  = (h == 0) ? 1.0f : (1.0f + 7.0f*sigm(xp[4+h]));
        float ampf = sigm(xp[164+h]); ampf *= ampf;
        float m0   = sigm(xp[12+h])*0.9999f; m0 *= m0;
        if (tid < NF) mgf[tid] = powf(m0, (float)(tid+1));

        float run = 0.f;
        for (int s = 0; s < 32; ++s) {
            int t = base + s;
            float f = MINF + lin_interp_arr(f0fr, NF, t, NS)*FSPAN;
            if (f > 1.0f) f = 0.f;
            run += f*fac*PI_F;
        }
        float off = block_excl_scan_1024(run, wsum);
        run = 0.f;
        for (int s = 0; s < 32; ++s) {
            int t = base + s;
            float f = MINF + lin_interp_arr(f0fr, NF, t, NS)*FSPAN;
            if (f > 1.0f) f = 0.f;
            run += f*fac*PI_F;
            float osc = sinf(off + run)*ampf;
            float mg  = lin_interp_arr(mgf, NF, t, NS);
            acc[s] += osc*mg;
        }
        __syncthreads();
    }
    float* rp = resbuf + se*NS + base;
    #pragma unroll
    for (int s = 0; s < 32; ++s) rp[s] = acc[s];
}

// ---------------------------------------------------------------------------
// Kernel 3: forward conv FFTs. c = a_half + i*r_half (zero-padded to 32768),
// C^2 accumulated per batch in scrambled order (e3: 0->P=a0r0, 1->Q+=a0r1, 2->Q+=a1r0)
// ---------------------------------------------------------------------------
extern "C" __global__ void __launch_bounds__(512)
k_conv_fwd(const float* __restrict__ atoms, const float* __restrict__ res,
           float* __restrict__ accP, float* __restrict__ accQ)
{
    extern __shared__ float smem[];
    float* re = smem;
    float* im = smem + NS;
    const int bid = blockIdx.x;
    const int e3  = bid % 3;
    const int se  = bid / 3;
    const int b   = se >> 3;
    const float* A = atoms + se*NS;
    const float* R = res   + se*NS;
    const float* ah = (e3 == 2) ? (A + HALF_NS) : A;
    const float* rh = (e3 == 1) ? (R + HALF_NS) : R;
    const int tid = threadIdx.x;

#ifdef HAVE_TDM
    if (tid < 32) {
        tdm_load_1d(re, ah, HALF_NS);
        tdm_load_1d(im, rh, HALF_NS);
        __builtin_amdgcn_s_wait_tensorcnt(0);
    }
#else
    for (int t = tid; t < HALF_NS; t += blockDim.x) { re[t] = ah[t]; im[t] = rh[t]; }
#endif
    for (int t = HALF_NS + tid; t < NS; t += blockDim.x) { re[t] = 0.f; im[t] = 0.f; }
    __syncthreads();
    fft32k(re, im, false);

    float* accR = ((e3 == 0) ? accP : accQ) + b*(2*NS);
    float* accI = accR + NS;
    for (int t = tid; t < NS; t += blockDim.x) {
        float cr = re[t], ci = im[t];
        unsafeAtomicAdd(&accR[t], cr*cr - ci*ci);
        unsafeAtomicAdd(&accI[t], 2.0f*cr*ci);
    }
}

// ---------------------------------------------------------------------------
// Kernel 4/5: inverse FFT of accumulated spectra -> imag/2 is the conv sum.
// P writes all of d_out; Q (launched after) adds shifted-by-16384 part.
// ---------------------------------------------------------------------------
extern "C" __global__ void __launch_bounds__(512)
k_conv_inv_p(const float* __restrict__ accP, float* __restrict__ out)
{
    extern __shared__ float smem[];
    float* re = smem;
    float* im = smem + NS;
    const int b = blockIdx.x;
    const float* aR = accP + b*(2*NS);
    const float* aI = aR + NS;
    const int tid = threadIdx.x;
#ifdef HAVE_TDM
    if (tid < 32) {
        tdm_load_1d(re, aR, NS);
        tdm_load_1d(im, aI, NS);
        __builtin_amdgcn_s_wait_tensorcnt(0);
    }
#else
    for (int t = tid; t < NS; t += blockDim.x) { re[t] = aR[t]; im[t] = aI[t]; }
#endif
    __syncthreads();
    fft32k(re, im, true);
    const float sc = 0.5f/(float)NS;
    for (int t = tid; t < NS; t += blockDim.x) out[b*NS + t] = im[t]*sc;
}

extern "C" __global__ void __launch_bounds__(512)
k_conv_inv_q(const float* __restrict__ accQ, float* __restrict__ out)
{
    extern __shared__ float smem[];
    float* re = smem;
    float* im = smem + NS;
    const int b = blockIdx.x;
    const float* aR = accQ + b*(2*NS);
    const float* aI = aR + NS;
    const int tid = threadIdx.x;
#ifdef HAVE_TDM
    if (tid < 32) {
        tdm_load_1d(re, aR, NS);
        tdm_load_1d(im, aI, NS);
        __builtin_amdgcn_s_wait_tensorcnt(0);
    }
#else
    for (int t = tid; t < NS; t += blockDim.x) { re[t] = aR[t]; im[t] = aI[t]; }
#endif
    __syncthreads();
    fft32k(re, im, true);
    const float sc = 0.5f/(float)NS;
    for (int t = tid; t < HALF_NS; t += blockDim.x) out[b*NS + HALF_NS + t] += im[t]*sc;
}

extern "C" __global__ void k_zero(float* __restrict__ p, int n)
{
    int i = blockIdx.x*blockDim.x + threadIdx.x;
    if (i < n) p[i] = 0.f;
}

// ---------------------------------------------------------------------------
extern "C" void kernel_launch(void* const* d_in, const int* in_sizes, int n_in,
                              void* d_out, int out_size, void* d_ws, size_t ws_size,
                              hipStream_t stream)
{
    const float* x     = (const float*)d_in[0];
    const float* noise = (const float*)d_in[1];
    float* out = (float*)d_out;
    float* ws  = (float*)d_ws;

    float* atoms = ws;                              // 128*32768
    float* resb  = ws + 128*NS;                     // 128*32768
    float* accP  = ws + 2*128*NS;                   // 16*2*32768
    float* accQ  = accP + NB*2*NS;                  // 16*2*32768

    const int SMEM = 2*NS*(int)sizeof(float) + 1024;
    (void)hipFuncSetAttribute((const void*)k_atoms,      hipFuncAttributeMaxDynamicSharedMemorySize, SMEM);
    (void)hipFuncSetAttribute((const void*)k_conv_fwd,   hipFuncAttributeMaxDynamicSharedMemorySize, SMEM);
    (void)hipFuncSetAttribute((const void*)k_conv_inv_p, hipFuncAttributeMaxDynamicSharedMemorySize, SMEM);
    (void)hipFuncSetAttribute((const void*)k_conv_inv_q, hipFuncAttributeMaxDynamicSharedMemorySize, SMEM);

    const int nAcc = NB*2*NS*2;                     // accP + accQ elements
    k_zero      <<<(nAcc + 1023)/1024, 1024, 0, stream>>>(accP, nAcc);
    k_atoms     <<<128, 512, SMEM, stream>>>(x, noise, atoms);
    k_res       <<<128, 1024, 0, stream>>>(x, resb);
    k_conv_fwd  <<<384, 512, SMEM, stream>>>(atoms, resb, accP, accQ);
    k_conv_inv_p<<<NB, 512, SMEM, stream>>>(accP, out);
    k_conv_inv_q<<<NB, 512, SMEM, stream>>>(accQ, out);
}